// AnalysisLayer_82995948028301
// MI455X (gfx1250) — compile-verified
//
#include <hip/hip_runtime.h>
#include <hip/hip_bf16.h>

// ---------------------------------------------------------------------------
// AnalysisLayer: stream-copy data_in -> out while reducing per-channel
// mean(|x|) over (B,H,W); energy_out = energy_in + mean; hist_out = hist_in+1.
//
// B=32, C=256, H=W=64.  Element index = ((b*C + c)*H*W) + i, so each
// contiguous 4096-float slab belongs to exactly one channel -> one block per
// slab, block id = b*C + c.  Memory-bound: ~256 MB traffic, ~11 us @23.3TB/s.
// Wave-level reduction uses V_WMMA_F32_16X16X4_F32 with an all-ones B matrix
// (row-sum trick), wave32 layouts per CDNA5 ISA.
// ---------------------------------------------------------------------------

typedef __attribute__((ext_vector_type(2))) float v2f;
typedef __attribute__((ext_vector_type(4))) float f4v;
typedef __attribute__((ext_vector_type(8))) float v8f;

#define B_DIM 32
#define C_DIM 256
#define HW    4096                  // 64*64, one block's slab
#define NDATA ((size_t)B_DIM * C_DIM * HW)
#define HIST_N 10000

__global__ __launch_bounds__(256)
void reduce_copy_kernel(const float* __restrict__ in,
                        float* __restrict__ out,
                        float* __restrict__ ws) {
    __shared__ float wsum[8];

    const int  blk  = blockIdx.x;                 // == b*C + c
    const size_t base = (size_t)blk * HW;
    const int  t    = threadIdx.x;

    const f4v* __restrict__ in4  = (const f4v*)(in  + base);
    f4v*       __restrict__ out4 = (f4v*)      (out + base);

    // 256 threads * 4 float4 = 4096 floats. Streaming: NT load + NT store.
    float s = 0.0f;
#pragma unroll
    for (int i = 0; i < 4; ++i) {
        f4v v = __builtin_nontemporal_load(&in4[t + i * 256]);
        __builtin_nontemporal_store(v, &out4[t + i * 256]);
        s += __builtin_fabsf(v.x) + __builtin_fabsf(v.y) +
             __builtin_fabsf(v.z) + __builtin_fabsf(v.w);
    }

    // ---- wave32 reduction via WMMA row-sum trick -------------------------
    // A (16x4 f32, 2 VGPRs): lanes 0..15 supply K=0 (VGPR0) / K=1 (VGPR1),
    // lanes 16..31 supply K=2 / K=3.  a0 = per-lane partial, a1 = 0
    //   => A[m][0] = s(lane m), A[m][2] = s(lane m+16), K=1,3 are zero.
    // B = all ones (both B VGPRs = 1.0) => D[m][n] = s(m) + s(m+16).
    v2f a;  a[0] = s;    a[1] = 0.0f;
    v2f b1; b1[0] = 1.0f; b1[1] = 1.0f;
    v8f c = {};
    c = __builtin_amdgcn_wmma_f32_16x16x4_f32(
            /*neg_a=*/false, a, /*neg_b=*/false, b1,
            /*c_mod=*/(short)0, c, /*reuse_a=*/false, /*reuse_b=*/false);

    // D layout: VGPR i holds row M=i (lanes 0..15) and M=8+i (lanes 16..31).
    // Summing the 8 D VGPRs gives rows 0..7 in the low half-lanes and rows
    // 8..15 in the high half; xor-16 completes the full wave sum.
    float tsum = c[0] + c[1] + c[2] + c[3] + c[4] + c[5] + c[6] + c[7];
    tsum += __shfl_xor(tsum, 16, 32);

    const int wave = t >> 5;
    if ((t & 31) == 0) wsum[wave] = tsum;
    __syncthreads();

    if (t == 0) {
        float tot = 0.0f;
#pragma unroll
        for (int w = 0; w < 8; ++w) tot += wsum[w];
        ws[blk] = tot;              // per-(b,c) block sum of |x|
    }
}

// Fold 32 batch-partials per channel; deterministic serial order.
__global__ __launch_bounds__(256)
void finalize_energy_kernel(const float* __restrict__ ws,
                            const float* __restrict__ energy_in,
                            float* __restrict__ energy_out) {
    const int ch = threadIdx.x;     // 256 threads, 1 block
    float s = 0.0f;
#pragma unroll
    for (int b = 0; b < B_DIM; ++b) s += ws[b * C_DIM + ch];
    energy_out[ch] = energy_in[ch] + s * (1.0f / (float)(B_DIM * HW));
}

__global__ __launch_bounds__(256)
void hist_kernel(const float* __restrict__ hin,
                 float* __restrict__ hout, int n) {
    int i = blockIdx.x * blockDim.x + threadIdx.x;
    if (i < n) hout[i] = hin[i] + 1.0f;
}

extern "C" void kernel_launch(void* const* d_in, const int* in_sizes, int n_in,
                              void* d_out, int out_size, void* d_ws, size_t ws_size,
                              hipStream_t stream) {
    const float* data_in   = (const float*)d_in[0];
    const float* energy_in = (const float*)d_in[1];
    const float* hist_in   = (const float*)d_in[2];

    float* out        = (float*)d_out;
    float* energy_out = out + NDATA;          // outputs concatenated flat
    float* hist_out   = energy_out + C_DIM;
    float* ws         = (float*)d_ws;          // 8192 floats of scratch

    reduce_copy_kernel<<<B_DIM * C_DIM, 256, 0, stream>>>(data_in, out, ws);
    finalize_energy_kernel<<<1, C_DIM, 0, stream>>>(ws, energy_in, energy_out);
    hist_kernel<<<(HIST_N + 255) / 256, 256, 0, stream>>>(hist_in, hist_out, HIST_N);
}